// RFPEncoding_33552284516500
// MI455X (gfx1250) — compile-verified
//
#include <hip/hip_runtime.h>

typedef __attribute__((ext_vector_type(2))) float v2f;
typedef __attribute__((ext_vector_type(8))) float v8f;

#define N_NODES 1024
#define DFEAT   1024
#define BATCH   8
#define LAYERS  4
#define NEDGE   (N_NODES * 16)
#define M2      (N_NODES * DFEAT)      // elements per 1024x1024 matrix
#define NS_ITERS 12

// ---------------------------------------------------------------------------
// gfx1250 async global->LDS helpers (ASYNCcnt-tracked, per cdna5 ISA ch.10 / 15.18)
// LDS destination offset = low 32 bits of the flat shared-aperture address
// (ISA 10.2).
// ---------------------------------------------------------------------------
__device__ __forceinline__ unsigned lds_off(const void* p) {
    return (unsigned)(unsigned long long)p;
}
__device__ __forceinline__ void async_copy_b128(const float* gptr, unsigned ldsoff) {
    asm volatile("global_load_async_to_lds_b128 %0, %1, off"
                 :: "v"(ldsoff), "v"(gptr) : "memory");
}
template <int N>
__device__ __forceinline__ void wait_asynccnt() {
    asm volatile("s_wait_asynccnt %0" :: "i"(N) : "memory");
}

// ---------------------------------------------------------------------------
// Edge-range lookup: edge list is sorted by src (src = repeat(arange(N),16)).
// Binary search gives a deterministic, atomic-free segment sum.
// ---------------------------------------------------------------------------
__device__ __forceinline__ int lower_bound_src(const int* __restrict__ ei, int val) {
    int lo = 0, hi = NEDGE;
    while (lo < hi) {
        int mid = (lo + hi) >> 1;
        if (ei[2 * mid] < val) lo = mid + 1; else hi = mid;
    }
    return lo;
}

__global__ void k_degree(const int* __restrict__ ei, float* __restrict__ deg) {
    int i = blockIdx.x * blockDim.x + threadIdx.x;
    if (i < N_NODES) {
        int lo = lower_bound_src(ei, i);
        int hi = lower_bound_src(ei, i + 1);
        deg[i] = (float)(hi - lo);
    }
}

// agg[b,i,d] = 0.5*deg_i*x[b,i,d] + sum_{e: src=i} 0.5/sqrt(deg_i*deg_dst) * x[b,dst,d]
__global__ void k_aggregate(const float* __restrict__ x, const int* __restrict__ ei,
                            const float* __restrict__ deg, float* __restrict__ agg) {
    int d = blockIdx.x * blockDim.x + threadIdx.x;
    int i = blockIdx.y;
    int b = blockIdx.z;
    int lo = lower_bound_src(ei, i);
    int hi = lower_bound_src(ei, i + 1);
    float di  = deg[i];
    float acc = 0.5f * di * x[(size_t)b * M2 + i * DFEAT + d];   // mean term
    float wsi = 0.5f * rsqrtf(fmaxf(di, 1.0f));
    for (int e = lo; e < hi; ++e) {
        int t = ei[2 * e + 1];
        float w = wsi * rsqrtf(fmaxf(deg[t], 1.0f));
        acc += w * x[(size_t)b * M2 + t * DFEAT + d];
    }
    agg[(size_t)b * M2 + i * DFEAT + d] = acc;
}

// ---------------------------------------------------------------------------
// trace(G) per batch -> s[b]  (used to normalize Y0 = G / tr(G))
// ---------------------------------------------------------------------------
__global__ void k_trace(const float* __restrict__ G, float* __restrict__ s) {
    __shared__ float red[256];
    int b = blockIdx.x;
    float acc = 0.f;
    for (int i = threadIdx.x; i < N_NODES; i += 256)
        acc += G[(size_t)b * M2 + i * DFEAT + i];
    red[threadIdx.x] = acc;
    __syncthreads();
    for (int st = 128; st > 0; st >>= 1) {
        if (threadIdx.x < st) red[threadIdx.x] += red[threadIdx.x + st];
        __syncthreads();
    }
    if (threadIdx.x == 0) s[b] = fmaxf(red[0], 1e-20f);
}

__global__ void k_scale_inplace(float* __restrict__ Y, const float* __restrict__ s) {
    int i = blockIdx.x * blockDim.x + threadIdx.x;
    int b = blockIdx.z;
    Y[(size_t)b * M2 + i] *= (1.0f / s[b]);
}

__global__ void k_set_identity(float* __restrict__ Z) {
    int d = blockIdx.x * blockDim.x + threadIdx.x;
    int n = blockIdx.y;
    int b = blockIdx.z;
    Z[(size_t)b * M2 + n * DFEAT + d] = (n == d) ? 1.0f : 0.0f;
}

__global__ void k_writeout(const float* __restrict__ src, float* __restrict__ out, int layer) {
    int d = blockIdx.x * blockDim.x + threadIdx.x;
    int n = blockIdx.y;
    int b = blockIdx.z;
    out[((size_t)(b * N_NODES + n)) * (DFEAT * (LAYERS + 1)) + layer * DFEAT + d] =
        src[(size_t)b * M2 + n * DFEAT + d];
}

// ---------------------------------------------------------------------------
// WMMA f32 GEMM:  C = alpha_eff * op(A) * B + beta * I     (TB == 0 always)
//   TA template: 1 -> op(A) = A^T (used for G = A^T A), 0 -> op(A) = A.
//   smode: 0 -> alpha_eff = alpha
//          1 -> alpha_eff = alpha / s[b]
//          2 -> alpha_eff = alpha * rsqrt(s[b])
//
// Block: 256 threads (8 waves). Block tile 128(M) x 64(N); wave tile 32x32
// (2x2 of 16x16 WMMA tiles); K staged through double-buffered LDS in slabs
// of 32 via GLOBAL_LOAD_ASYNC_TO_LDS_B128 (6 async ops per wave per stage:
// A slab 128x32 -> 4/thread, B slab 32x64 -> 2/thread). After issuing stage
// i+1, s_wait_asynccnt 6 guarantees stage i is resident (async loads retire
// in order), overlapping the next fetch with the 32 WMMAs of compute.
//
// LDS layouts follow global contiguity (async op copies 16 contiguous bytes):
//   TA=1: As[k][TM+4]  (contiguous in m; row stride 132 dwords, 16B aligned)
//   TA=0: As[m][KS+4]  (contiguous in k; row stride 36 dwords -> 16B aligned
//                       and 16 distinct banks across lanes for fragment reads)
//   B   : Bs[k][TN+4]  (contiguous in n; row stride 68 dwords)
//
// Fragment layouts per CDNA5 ISA 7.12.2 (f32 16x16x4):
//   A: lane L(0..15) row M=L holds K=klo,klo+1 ; klo=(lane>>4)*2 ; v2f/lane
//   B: lane L(0..15) col N=L holds K=klo,klo+1
//   C/D: vgpr r, lanes 0..15 -> M=r ; lanes 16..31 -> M=r+8 ; N=lane&15
// ---------------------------------------------------------------------------
#define TM 128
#define TN 64
#define KS 32

template <int TA>
__global__ __launch_bounds__(256)
void k_gemm_wmma(const float* __restrict__ A, const float* __restrict__ B,
                 float* __restrict__ C, float alpha, float beta,
                 const float* __restrict__ sptr, int smode)
{
    __shared__ float As[2][TA ? (KS * (TM + 4)) : (TM * (KS + 4))];
    __shared__ float Bs[2][KS * (TN + 4)];

    const int b    = blockIdx.z;
    const int tid  = threadIdx.x;
    const int lane = tid & 31;
    const int wave = tid >> 5;        // 0..7
    const int wm   = wave & 3;        // 4 waves tiling M
    const int wn   = wave >> 2;       // 2 waves tiling N
    const int m0   = blockIdx.y * TM;
    const int n0   = blockIdx.x * TN;

    const float* __restrict__ Ab = A + (size_t)b * M2;
    const float* __restrict__ Bb = B + (size_t)b * M2;
    float* __restrict__       Cb = C + (size_t)b * M2;

    v8f acc[2][2];
    acc[0][0] = v8f{}; acc[0][1] = v8f{};
    acc[1][0] = v8f{}; acc[1][1] = v8f{};

    const int klo = (lane >> 4) << 1;   // 0 or 2
    const int l15 = lane & 15;

    // Issue one K-slab's async copies into LDS buffer `buf`.
    auto issue_stage = [&](int k0, int buf) {
        #pragma unroll
        for (int j = 0; j < 4; ++j) {            // A slab: 1024 16B chunks
            int c = tid + 256 * j;
            if (TA) {
                int k  = c >> 5;                 // 32 chunks per k-row of 128
                int m4 = (c & 31) << 2;
                const float* g = Ab + (size_t)(k0 + k) * DFEAT + m0 + m4;
                async_copy_b128(g, lds_off(&As[buf][k * (TM + 4) + m4]));
            } else {
                int m  = c >> 3;                 // 8 chunks per m-row of 32
                int k4 = (c & 7) << 2;
                const float* g = Ab + (size_t)(m0 + m) * DFEAT + k0 + k4;
                async_copy_b128(g, lds_off(&As[buf][m * (KS + 4) + k4]));
            }
        }
        #pragma unroll
        for (int j = 0; j < 2; ++j) {            // B slab: 512 16B chunks
            int c  = tid + 256 * j;
            int k  = c >> 4;                     // 16 chunks per k-row of 64
            int n4 = (c & 15) << 2;
            const float* g = Bb + (size_t)(k0 + k) * DFEAT + n0 + n4;
            async_copy_b128(g, lds_off(&Bs[buf][k * (TN + 4) + n4]));
        }
    };

    const int NST = DFEAT / KS;   // 32 stages
    issue_stage(0, 0);

    for (int st = 0; st < NST; ++st) {
        const int buf = st & 1;
        if (st + 1 < NST) {
            issue_stage((st + 1) * KS, buf ^ 1);
            wait_asynccnt<6>();   // newest 6 (next stage) may remain in flight
        } else {
            wait_asynccnt<0>();
        }
        __syncthreads();

        const float* __restrict__ Asb = &As[buf][0];
        const float* __restrict__ Bsb = &Bs[buf][0];

        #pragma unroll
        for (int kk = 0; kk < KS; kk += 4) {
            v2f afr[2], bfr[2];
            #pragma unroll
            for (int sm = 0; sm < 2; ++sm) {
                int m = wm * 32 + sm * 16 + l15;
                if (TA) {
                    afr[sm].x = Asb[(kk + klo + 0) * (TM + 4) + m];
                    afr[sm].y = Asb[(kk + klo + 1) * (TM + 4) + m];
                } else {
                    afr[sm].x = Asb[m * (KS + 4) + kk + klo + 0];
                    afr[sm].y = Asb[m * (KS + 4) + kk + klo + 1];
                }
            }
            #pragma unroll
            for (int sn = 0; sn < 2; ++sn) {
                int n = wn * 32 + sn * 16 + l15;
                bfr[sn].x = Bsb[(kk + klo + 0) * (TN + 4) + n];
                bfr[sn].y = Bsb[(kk + klo + 1) * (TN + 4) + n];
            }
            #pragma unroll
            for (int sm = 0; sm < 2; ++sm)
                #pragma unroll
                for (int sn = 0; sn < 2; ++sn)
                    acc[sm][sn] = __builtin_amdgcn_wmma_f32_16x16x4_f32(
                        /*neg_a=*/false, afr[sm], /*neg_b=*/false, bfr[sn],
                        /*c_mod=*/(short)0, acc[sm][sn],
                        /*reuse_a=*/false, /*reuse_b=*/false);
        }
        __syncthreads();
    }

    float aeff = alpha;
    if (smode == 1)      aeff = alpha / sptr[b];
    else if (smode == 2) aeff = alpha * rsqrtf(sptr[b]);

    const int rhalf = (lane >> 4) << 3;   // 0 or 8
    #pragma unroll
    for (int sm = 0; sm < 2; ++sm) {
        #pragma unroll
        for (int sn = 0; sn < 2; ++sn) {
            int Mbase = m0 + wm * 32 + sm * 16 + rhalf;
            int Nc    = n0 + wn * 32 + sn * 16 + l15;
            #pragma unroll
            for (int r = 0; r < 8; ++r) {
                int Mr = Mbase + r;
                float vv = aeff * acc[sm][sn][r];
                if (Mr == Nc) vv += beta;
                Cb[(size_t)Mr * DFEAT + Nc] = vv;
            }
        }
    }
}

// ---------------------------------------------------------------------------
// Host-side orchestration (graph-capture safe: launches only, all on stream)
// ---------------------------------------------------------------------------
extern "C" void kernel_launch(void* const* d_in, const int* in_sizes, int n_in,
                              void* d_out, int out_size, void* d_ws, size_t ws_size,
                              hipStream_t stream) {
    (void)in_sizes; (void)n_in; (void)out_size; (void)ws_size;

    const float* x0 = (const float*)d_in[0];
    const int*   ei = (const int*)d_in[1];
    float*       out = (float*)d_out;

    char*  ws  = (char*)d_ws;
    size_t big = (size_t)BATCH * M2 * sizeof(float);   // 32 MB per batched matrix

    float* Xbuf0 = (float*)(ws + 0 * big);
    float* Xbuf1 = (float*)(ws + 1 * big);
    float* Aagg  = (float*)(ws + 2 * big);
    float* Y     = (float*)(ws + 3 * big);
    float* Z     = (float*)(ws + 4 * big);
    float* T     = (float*)(ws + 5 * big);
    float* U     = (float*)(ws + 6 * big);
    float* V     = (float*)(ws + 7 * big);
    float* deg   = (float*)(ws + 8 * big);
    float* sc    = deg + N_NODES;

    dim3 blk(256);
    dim3 gridND(DFEAT / 256, N_NODES, BATCH);             // (4, 1024, 8)
    dim3 gridGemm(DFEAT / TN, N_NODES / TM, BATCH);       // (16, 8, 8)

    k_degree<<<dim3((N_NODES + 255) / 256), blk, 0, stream>>>(ei, deg);
    k_writeout<<<gridND, blk, 0, stream>>>(x0, out, 0);

    const float* xcur = x0;
    int cur = 0;

    for (int l = 0; l < LAYERS; ++l) {
        // A = agg(xcur)
        k_aggregate<<<gridND, blk, 0, stream>>>(xcur, ei, deg, Aagg);

        // G = A^T A  -> Y ; s = tr(G) ; Y /= s ; Z = I
        k_gemm_wmma<1><<<gridGemm, blk, 0, stream>>>(Aagg, Aagg, Y, 1.f, 0.f, sc, 0);
        k_trace<<<dim3(BATCH), blk, 0, stream>>>(Y, sc);
        k_scale_inplace<<<dim3(M2 / 256, 1, BATCH), blk, 0, stream>>>(Y, sc);
        k_set_identity<<<gridND, blk, 0, stream>>>(Z);

        // Newton–Schulz: T = 1.5I - 0.5 Z Y ; Y <- Y T ; Z <- T Z
        for (int it = 0; it < NS_ITERS; ++it) {
            k_gemm_wmma<0><<<gridGemm, blk, 0, stream>>>(Z, Y, T, -0.5f, 1.5f, sc, 0);
            k_gemm_wmma<0><<<gridGemm, blk, 0, stream>>>(Y, T, U, 1.0f, 0.0f, sc, 0);
            k_gemm_wmma<0><<<gridGemm, blk, 0, stream>>>(T, Z, V, 1.0f, 0.0f, sc, 0);
            float* t1 = Y; Y = U; U = t1;
            float* t2 = Z; Z = V; V = t2;
        }

        // Q = A * Z * rsqrt(s)  -> next x
        float* xnext = (cur == 0) ? Xbuf0 : Xbuf1;
        k_gemm_wmma<0><<<gridGemm, blk, 0, stream>>>(Aagg, Z, xnext, 1.0f, 0.0f, sc, 2);
        k_writeout<<<gridND, blk, 0, stream>>>(xnext, out, l + 1);

        xcur = xnext;
        cur ^= 1;
    }
}